// _IntxWeightQuantizedEmbedding_1812476199313
// MI455X (gfx1250) — compile-verified
//
#include <hip/hip_runtime.h>
#include <stdint.h>

// Quantized-embedding gather + per-group dequant for MI455X (gfx1250).
// Bandwidth-bound: ~83 MB total traffic -> ~3.6us floor at 23.3 TB/s.
// Uses the CDNA5 async global->LDS engine (GLOBAL_LOAD_ASYNC_TO_LDS_B128,
// ASYNCcnt) to stage gathered 1KB rows, double-buffered per wave.

#define DIM      1024
#define NG       8          // groups per row (GROUP=128)
#define WPB      8          // waves per block (256 threads, wave32)
#define ROW_B    1024       // bytes of qvals per token row
#define SMEM_B   (WPB * 2 * ROW_B)   // double buffer per wave = 16 KB/block
#define TPW      4          // tokens per wave (grid sizing)

// Issue async copy of one 1KB row into LDS. Each lane moves 16B; two B128
// async ops cover 1024B. Per ISA 15.18, the instruction offset is added to
// BOTH the LDS address (VDST vgpr) and the global address, so offset:512
// handles the second half with the same address VGPRs.
__device__ __forceinline__ void async_row_to_lds(const signed char* gbase,
                                                 uint32_t lds_va, uint32_t voff) {
  asm volatile(
      "global_load_async_to_lds_b128 %0, %1, %2 offset:0\n"
      "global_load_async_to_lds_b128 %0, %1, %2 offset:512"
      :
      : "v"(lds_va), "v"(voff), "s"((uint64_t)(uintptr_t)gbase)
      : "memory");
}

__global__ __launch_bounds__(256) void dequant_embed_kernel(
    const signed char* __restrict__ qvals,   // [NUM_EMB, DIM] int8 (int4 values)
    const float*       __restrict__ scales,  // [NUM_EMB, NG]  f32
    const signed char* __restrict__ zeros,   // [NUM_EMB, NG]  int8
    const int*         __restrict__ x,       // [T] token ids
    float*             __restrict__ out,     // [T, DIM] f32
    int T, int W) {
  __shared__ __align__(16) uint8_t smem[SMEM_B];

  const int lane = threadIdx.x & 31;
  const int wib  = threadIdx.x >> 5;                       // wave in block
  const int wid  = __builtin_amdgcn_readfirstlane((int)(blockIdx.x * WPB + wib));

  uint8_t* b0 = smem + (size_t)(wib * 2) * ROW_B;
  uint8_t* b1 = b0 + ROW_B;
  // Low 32 bits of a generic LDS pointer == LDS byte offset (ISA 10.2).
  const uint32_t base0 = (uint32_t)(uintptr_t)b0;
  const uint32_t base1 = (uint32_t)(uintptr_t)b1;
  const uint32_t voff  = (uint32_t)lane * 16u;

  int t = wid;
  if (t >= T) return;   // wave-uniform exit

  // Prologue: fetch first row into buffer 0.
  int row = __builtin_amdgcn_readfirstlane(x[t]);
  async_row_to_lds(qvals + (size_t)row * DIM, base0 + voff, voff);

  int p = 0;
  for (;;) {
    // Prefetch next token's row into the other buffer (clamped dummy refetch
    // on the last iteration so the in-flight count stays constant).
    const int tn = t + W;
    const int tf = (tn < T) ? tn : t;
    const int row_next = __builtin_amdgcn_readfirstlane(x[tf]);
    async_row_to_lds(qvals + (size_t)row_next * DIM,
                     (p ? base0 : base1) + voff, voff);

    // 4 async ops may be outstanding; loads complete in order, so
    // ASYNCcnt<=2 guarantees the current buffer's two ops are done.
    asm volatile("s_wait_asynccnt 0x2" ::: "memory");

    // Dequantize current buffer. Group k is wave-uniform -> scalar s/z loads.
    const uint32_t*    lp   = (const uint32_t*)(p ? b1 : b0);
    const float*       srow = scales + (size_t)row * NG;
    const signed char* zrow = zeros  + (size_t)row * NG;
    float4*            op   = (float4*)(out + (size_t)t * DIM);
#pragma unroll
    for (int k = 0; k < NG; ++k) {
      const float s   = srow[k];
      const float nzs = -(float)zrow[k] * s;          // fold zero into fma addend
      const uint32_t q = lp[k * 32 + lane];           // ds_load_b32, no bank conflicts
      float4 v;
      v.x = fmaf((float)((int)(q << 24) >> 24), s, nzs);
      v.y = fmaf((float)((int)(q << 16) >> 24), s, nzs);
      v.z = fmaf((float)((int)(q <<  8) >> 24), s, nzs);
      v.w = fmaf((float)((int)q >> 24),         s, nzs);
      op[k * 32 + lane] = v;                          // coalesced global_store_b128
    }

    if (tn >= T) break;       // trailing dummy async drains at s_endpgm (wait-idle)
    t = tn;
    row = row_next;
    p ^= 1;
  }
}

extern "C" void kernel_launch(void* const* d_in, const int* in_sizes, int n_in,
                              void* d_out, int out_size, void* d_ws, size_t ws_size,
                              hipStream_t stream) {
  const signed char* qvals  = (const signed char*)d_in[0];
  const float*       scales = (const float*)d_in[1];
  const signed char* zeros  = (const signed char*)d_in[2];
  const int*         x      = (const int*)d_in[3];
  float*             out    = (float*)d_out;

  const int T = in_sizes[3];                       // number of tokens (16384)
  int blocks = (T + WPB * TPW - 1) / (WPB * TPW);  // 4 tokens per wave
  if (blocks < 1) blocks = 1;
  const int W = blocks * WPB;                      // total waves (grid stride)

  dequant_embed_kernel<<<blocks, WPB * 32, 0, stream>>>(qvals, scales, zeros, x,
                                                        out, T, W);
}